// TTRCell_32813550141485
// MI455X (gfx1250) — compile-verified
//
#include <hip/hip_runtime.h>

// CDNA5 / gfx1250: wave32, WMMA 16x16x4 F32 (D = A*B + C, f32 throughout).
typedef float v2f __attribute__((ext_vector_type(2)));
typedef float v8f __attribute__((ext_vector_type(8)));

#define BH   64            // batch*heads
#define SEQ  4096          // sequence length
#define FD   64            // feature dim (K-dim of q,k)
#define VD   64            // value dim
#define CHK  64            // chunk length
#define NCH  (SEQ / CHK)   // 64 chunks
#define ST   68            // LDS row stride in floats (bank stride 4 -> conflict-free row reads)

__device__ __forceinline__ v8f wmma_f32_k4(v2f a, v2f b, v8f c) {
  // 8-arg form: (neg_a, A, neg_b, B, c_mod, C, reuse_a, reuse_b)
  return __builtin_amdgcn_wmma_f32_16x16x4_f32(false, a, false, b, (short)0, c,
                                               false, false);
}

// ---------------------------------------------------------------------------
// Pass 1: per (head, chunk) compute U = K_cᵀ V_c  (64x64 f32)  -> workspace
// ---------------------------------------------------------------------------
__global__ void __launch_bounds__(512)
ttr_chunk_kv(const float* __restrict__ Kp, const float* __restrict__ Vp,
             float* __restrict__ U) {
  __shared__ float sK[CHK * ST];
  __shared__ float sV[CHK * ST];
  const int ch = blockIdx.x, bh = blockIdx.y;
  const int tid = threadIdx.x;
  const size_t gbase = ((size_t)bh * SEQ + (size_t)ch * CHK) * FD;

#pragma unroll
  for (int p = 0; p < 2; ++p) {           // 2 float4 per thread per matrix
    const int i4 = tid + p * 512;
    const int e = i4 << 2;
    const int row = e >> 6, col = e & 63;
    const float4 kq = reinterpret_cast<const float4*>(Kp + gbase)[i4];
    const float4 vq = reinterpret_cast<const float4*>(Vp + gbase)[i4];
    *reinterpret_cast<float4*>(&sK[row * ST + col]) = kq;
    *reinterpret_cast<float4*>(&sV[row * ST + col]) = vq;
  }
  __syncthreads();

  const int lane = tid & 31, wv = tid >> 5;
  const int wi = wv >> 2, wj = wv & 3;    // 4x4 tile grid of 16x16 tiles
  const int lh = lane >> 4, ll = lane & 15;

  v8f acc = {};
#pragma unroll
  for (int k0 = 0; k0 < CHK; k0 += 4) {   // K-dim = time s within chunk
    const int kk = k0 + 2 * lh;
    v2f a, b;
    a.x = sK[(kk + 0) * ST + 16 * wi + ll];   // A[m,k] = K[s=k][f=16wi+m]  (Kᵀ)
    a.y = sK[(kk + 1) * ST + 16 * wi + ll];
    b.x = sV[(kk + 0) * ST + 16 * wj + ll];   // B[k,n] = V[s=k][v=16wj+n]
    b.y = sV[(kk + 1) * ST + 16 * wj + ll];
    acc = wmma_f32_k4(a, b, acc);
  }

  float* Ut = U + ((size_t)bh * NCH + ch) * (FD * VD);
#pragma unroll
  for (int r = 0; r < 8; ++r) {
    const int m = r + 8 * lh;
    Ut[(16 * wi + m) * VD + 16 * wj + ll] = acc[r];
  }
}

// ---------------------------------------------------------------------------
// Pass 2: per head, in-place exclusive prefix sum over chunks:
//         U[c] <- sum_{c'<c} U[c']   (64 x 4096 floats per head)
// ---------------------------------------------------------------------------
__global__ void __launch_bounds__(512)
ttr_scan(float* __restrict__ U) {
  const int bh = blockIdx.x, tid = threadIdx.x;
  float4* base = reinterpret_cast<float4*>(U + (size_t)bh * NCH * FD * VD);
  float4 a0 = make_float4(0.f, 0.f, 0.f, 0.f);
  float4 a1 = make_float4(0.f, 0.f, 0.f, 0.f);
  for (int c = 0; c < NCH; ++c) {
    float4* p = base + (size_t)c * (FD * VD / 4);
    if (c + 1 < NCH)
      __builtin_prefetch(p + (FD * VD / 4) + tid, 0, 0);  // global_prefetch_b8
    const float4 t0 = p[tid];
    const float4 t1 = p[tid + 512];
    p[tid] = a0;
    p[tid + 512] = a1;
    a0.x += t0.x; a0.y += t0.y; a0.z += t0.z; a0.w += t0.w;
    a1.x += t1.x; a1.y += t1.y; a1.z += t1.z; a1.w += t1.w;
  }
}

// ---------------------------------------------------------------------------
// Pass 3: per (head, chunk):
//   O_c = [ tril(Q_c K_cᵀ) V_c  +  Q_c W_c ] / (t+1)
// ---------------------------------------------------------------------------
__global__ void __launch_bounds__(512)
ttr_out(const float* __restrict__ Qp, const float* __restrict__ Kp,
        const float* __restrict__ Vp, const float* __restrict__ Wp,
        float* __restrict__ Op) {
  __shared__ float sQ[CHK * ST];
  __shared__ float sK[CHK * ST];
  __shared__ float sV[CHK * ST];
  __shared__ float sW[FD * ST];
  __shared__ float sP[CHK * ST];
  const int ch = blockIdx.x, bh = blockIdx.y;
  const int tid = threadIdx.x;
  const size_t gbase = ((size_t)bh * SEQ + (size_t)ch * CHK) * FD;
  const float* Wt = Wp + ((size_t)bh * NCH + ch) * (FD * VD);

#pragma unroll
  for (int p = 0; p < 2; ++p) {
    const int i4 = tid + p * 512;
    const int e = i4 << 2;
    const int row = e >> 6, col = e & 63;
    const float4 qq = reinterpret_cast<const float4*>(Qp + gbase)[i4];
    const float4 kq = reinterpret_cast<const float4*>(Kp + gbase)[i4];
    const float4 vq = reinterpret_cast<const float4*>(Vp + gbase)[i4];
    const float4 wq = reinterpret_cast<const float4*>(Wt)[i4];
    *reinterpret_cast<float4*>(&sQ[row * ST + col]) = qq;
    *reinterpret_cast<float4*>(&sK[row * ST + col]) = kq;
    *reinterpret_cast<float4*>(&sV[row * ST + col]) = vq;
    *reinterpret_cast<float4*>(&sW[row * ST + col]) = wq;
  }
  __syncthreads();

  const int lane = tid & 31, wv = tid >> 5;
  const int wi = wv >> 2, wj = wv & 3;
  const int lh = lane >> 4, ll = lane & 15;

  // S = Q K^T  (K-dim = feature)
  v8f s = {};
#pragma unroll
  for (int k0 = 0; k0 < FD; k0 += 4) {
    const int kk = k0 + 2 * lh;
    v2f a, b;
    a.x = sQ[(16 * wi + ll) * ST + kk + 0];   // A[m,k] = Q[t=16wi+m][f=k]
    a.y = sQ[(16 * wi + ll) * ST + kk + 1];
    b.x = sK[(16 * wj + ll) * ST + kk + 0];   // B[k,n] = Kᵀ[f=k][s=16wj+n]
    b.y = sK[(16 * wj + ll) * ST + kk + 1];
    s = wmma_f32_k4(a, b, s);
  }

  // causal mask (keep s <= t), spill P to LDS
#pragma unroll
  for (int r = 0; r < 8; ++r) {
    const int m = r + 8 * lh;
    const int t_loc = 16 * wi + m;
    const int s_loc = 16 * wj + ll;
    sP[t_loc * ST + s_loc] = (s_loc <= t_loc) ? s[r] : 0.0f;
  }
  __syncthreads();

  // O = P @ V  +  Q @ W   (one chained accumulator, 32 WMMAs)
  v8f o = {};
#pragma unroll
  for (int k0 = 0; k0 < CHK; k0 += 4) {       // intra-chunk: K-dim = time s
    const int kk = k0 + 2 * lh;
    v2f a, b;
    a.x = sP[(16 * wi + ll) * ST + kk + 0];
    a.y = sP[(16 * wi + ll) * ST + kk + 1];
    b.x = sV[(kk + 0) * ST + 16 * wj + ll];
    b.y = sV[(kk + 1) * ST + 16 * wj + ll];
    o = wmma_f32_k4(a, b, o);
  }
#pragma unroll
  for (int k0 = 0; k0 < FD; k0 += 4) {        // inter-chunk: K-dim = feature
    const int kk = k0 + 2 * lh;
    v2f a, b;
    a.x = sQ[(16 * wi + ll) * ST + kk + 0];
    a.y = sQ[(16 * wi + ll) * ST + kk + 1];
    b.x = sW[(kk + 0) * ST + 16 * wj + ll];
    b.y = sW[(kk + 1) * ST + 16 * wj + ll];
    o = wmma_f32_k4(a, b, o);
  }

#pragma unroll
  for (int r = 0; r < 8; ++r) {
    const int m = r + 8 * lh;
    const int tg = ch * CHK + 16 * wi + m;
    const float scale = 1.0f / (float)(tg + 1);
    Op[((size_t)bh * SEQ + tg) * VD + 16 * wj + ll] = o[r] * scale;
  }
}

// ---------------------------------------------------------------------------
extern "C" void kernel_launch(void* const* d_in, const int* in_sizes, int n_in,
                              void* d_out, int out_size, void* d_ws, size_t ws_size,
                              hipStream_t stream) {
  (void)in_sizes; (void)n_in; (void)out_size; (void)ws_size;
  const float* q = (const float*)d_in[0];
  const float* k = (const float*)d_in[1];
  const float* v = (const float*)d_in[2];
  float* U = (float*)d_ws;  // BH*NCH*FD*VD floats = 64 MiB chunk-state buffer

  dim3 grid(NCH, BH);       // 4096 workgroups for the WMMA passes
  dim3 block(512);          // 16 wave32s -> 4x4 grid of 16x16 tiles

  ttr_chunk_kv<<<grid, block, 0, stream>>>(k, v, U);
  ttr_scan<<<dim3(BH), block, 0, stream>>>(U);
  ttr_out<<<grid, block, 0, stream>>>(q, k, v, U, (float*)d_out);
}